// Attention_7241314861621
// MI455X (gfx1250) — compile-verified
//
#include <hip/hip_runtime.h>
#include <math.h>

#define NUM_HEADS 32
#define NUM_KV_HEADS 8
#define GROUP 4
#define HEAD_DIM 128
#define BS 16
#define KV_LEN 2048
#define ATTN_SCALE 0.08838834764831845f
#define NWAVES 2                             // waves per block (64 threads)
#define NTILES (KV_LEN / 16)                 // 128 key-tiles per (b, kvh)
#define NCHUNK 8                             // flash-decode KV split
#define PART_STRIDE (GROUP * (HEAD_DIM + 2)) // 520 floats per partial

typedef __attribute__((ext_vector_type(2))) float v2f;
typedef __attribute__((ext_vector_type(8))) float v8f;

__device__ __forceinline__ float wave_max(float v) {
#pragma unroll
    for (int off = 16; off; off >>= 1) v = fmaxf(v, __shfl_xor(v, off, 32));
    return v;
}
__device__ __forceinline__ float wave_sum(float v) {
#pragma unroll
    for (int off = 16; off; off >>= 1) v += __shfl_xor(v, off, 32);
    return v;
}

// Core: key-tiles [t0, t1) for one (b, kvh). Two waves stride over tiles with
// independent online softmax; K tiles are staged to LDS with double-buffered
// GLOBAL_LOAD_ASYNC_TO_LDS_B128 (ASYNCcnt), so HBM latency of tile t+1 hides
// behind the 64 f32 WMMAs of tile t. Wave partials merge through LDS.
// If write_partial: dst = 520-float partial entry {m[4], l[4], O[4][128]}.
// Else:             dst = out + (b, kvh) base, normalized store.
__device__ __forceinline__ void attn_core(
    int b, int kvh, int t0, int t1,
    const float* __restrict__ q, const float* __restrict__ kin,
    const float* __restrict__ vin, const float* __restrict__ kc,
    const float* __restrict__ vc, const int* __restrict__ slot_mapping,
    const int* __restrict__ page_indices,
    float (*sQ)[132], float (*sK)[2][16][132], float (*sP)[16 * 20],
    unsigned long long (*sVp)[2][16],
    float (*mM)[GROUP], float (*mL)[GROUP], float (*mO)[GROUP][HEAD_DIM],
    float* dst, bool write_partial)
{
    const int tid  = threadIdx.x;
    const int w    = tid >> 5;
    const int lane = tid & 31;
    const int half = (lane >> 4) & 1;
    const int l16  = lane & 15;

    // ---- stage Q: rows 0..3 = grouped query heads, rows 4..15 = 0 ----
    for (int i = tid; i < 16 * 132; i += NWAVES * 32) ((float*)sQ)[i] = 0.f;
    for (int i = lane; i < 16 * 20; i += 32) sP[w][i] = 0.f;
    __syncthreads();
    for (int i = tid; i < GROUP * HEAD_DIM; i += NWAVES * 32) {
        int h = i >> 7, d = i & 127;
        sQ[h][d] = q[(size_t)b * NUM_HEADS * HEAD_DIM + (kvh * GROUP + h) * HEAD_DIM + d];
    }
    __syncthreads();

    int sm[BS];
#pragma unroll
    for (int j = 0; j < BS; ++j) sm[j] = slot_mapping[j];

    // Issue one 16-key tile's K rows as async global->LDS B128 copies and
    // record the resolved (fresh-or-cache) V row pointers.
    auto stage = [&](int t, int bufi) {
#pragma unroll 4
        for (int r = 0; r < 16; ++r) {
            int lpos = t * 16 + r;
            int slot = page_indices[b * KV_LEN + lpos];
            int fresh = -1;
#pragma unroll
            for (int j = 0; j < BS; ++j)
                if (slot == sm[j]) fresh = j;
            const float* kp;
            const float* vp;
            if (fresh >= 0) {
                kp = kin + (size_t)fresh * NUM_KV_HEADS * HEAD_DIM + kvh * HEAD_DIM;
                vp = vin + (size_t)fresh * NUM_KV_HEADS * HEAD_DIM + kvh * HEAD_DIM;
            } else {
                kp = kc + (size_t)slot * NUM_KV_HEADS * HEAD_DIM + kvh * HEAD_DIM;
                vp = vc + (size_t)slot * NUM_KV_HEADS * HEAD_DIM + kvh * HEAD_DIM;
            }
            // Per-lane 16B: memory -> LDS, no VGPR round-trip, ASYNCcnt-tracked.
            unsigned ldsa = (unsigned)(unsigned long long)&sK[w][bufi][r][0]
                          + (unsigned)(lane * 16);
            const float* gp = kp + lane * 4;
            asm volatile("global_load_async_to_lds_b128 %0, %1, off"
                         :: "v"(ldsa), "v"(gp) : "memory");
            if (lane == 0) sVp[w][bufi][r] = (unsigned long long)vp;
        }
    };

    float m[GROUP], l[GROUP];
#pragma unroll
    for (int r = 0; r < GROUP; ++r) { m[r] = -INFINITY; l[r] = 0.f; }
    v8f acc[8];
    v8f vzero = {};
#pragma unroll
    for (int nt = 0; nt < 8; ++nt) acc[nt] = vzero;

    int t = t0 + w;
    if (t < t1) stage(t, 0);
    int buf = 0;
    for (; t < t1; t += NWAVES, buf ^= 1) {
        bool more = (t + NWAVES) < t1;
        if (more) {
            stage(t + NWAVES, buf ^ 1);  // prefetch next tile into other buffer
            // async loads retire in order: <=16 outstanding => current tile landed
            asm volatile("s_wait_asynccnt 16" ::: "memory");
        } else {
            asm volatile("s_wait_asynccnt 0" ::: "memory");
        }

        // ---- S = Q * K^T  (M=16 heads, N=16 keys, K=head_dim steps of 4) ----
        v8f c = vzero;
#pragma unroll 8
        for (int s = 0; s < 32; ++s) {
            int d0 = 4 * s + 2 * half;
            v2f a   = *(const v2f*)&sQ[l16][d0];
            v2f bfr = *(const v2f*)&sK[w][buf][l16][d0];
            c = __builtin_amdgcn_wmma_f32_16x16x4_f32(false, a, false, bfr,
                                                      (short)0, c, false, false);
        }

        // ---- online softmax per real head row ----
#pragma unroll
        for (int r = 0; r < GROUP; ++r) {
            float sv = (lane < 16) ? c[r] * ATTN_SCALE : -INFINITY;
            float nm = fmaxf(m[r], wave_max(sv));
            float p  = __expf(sv - nm);
            float tl = wave_sum(p);
            float fac = __expf(m[r] - nm);
            l[r] = l[r] * fac + tl;
            m[r] = nm;
            if (lane < 16) sP[w][r * 20 + lane] = p;
#pragma unroll
            for (int nt = 0; nt < 8; ++nt) acc[nt][r] *= fac;
        }

        // ---- O += P * V  (K=16 keys steps of 4, N=128 dims, 8 tiles) ----
#pragma unroll
        for (int ks = 0; ks < 4; ++ks) {
            int k0 = 4 * ks + 2 * half;
            v2f a = *(const v2f*)&sP[w][l16 * 20 + k0];
            const float* p0 = (const float*)sVp[w][buf][k0];
            const float* p1 = (const float*)sVp[w][buf][k0 + 1];
#pragma unroll
            for (int nt = 0; nt < 8; ++nt) {
                v2f bfr;
                bfr.x = p0[nt * 16 + l16];
                bfr.y = p1[nt * 16 + l16];
                acc[nt] = __builtin_amdgcn_wmma_f32_16x16x4_f32(false, a, false, bfr,
                                                                (short)0, acc[nt], false, false);
            }
        }
    }

    // ---- write per-wave partials, merge across the 2 waves ----
    if (lane < 16) {
#pragma unroll
        for (int nt = 0; nt < 8; ++nt)
#pragma unroll
            for (int r = 0; r < GROUP; ++r)
                mO[w][r][nt * 16 + lane] = acc[nt][r];
    }
    if (lane == 0) {
#pragma unroll
        for (int r = 0; r < GROUP; ++r) { mM[w][r] = m[r]; mL[w][r] = l[r]; }
    }
    __syncthreads();

    float gm[GROUP], f0[GROUP], f1[GROUP], Ls[GROUP];
#pragma unroll
    for (int r = 0; r < GROUP; ++r) {
        gm[r] = fmaxf(mM[0][r], mM[1][r]);
        f0[r] = __expf(mM[0][r] - gm[r]);
        f1[r] = __expf(mM[1][r] - gm[r]);
        Ls[r] = f0[r] * mL[0][r] + f1[r] * mL[1][r];
    }
    if (write_partial) {
        if (tid == 0) {
#pragma unroll
            for (int r = 0; r < GROUP; ++r) { dst[r] = gm[r]; dst[GROUP + r] = Ls[r]; }
        }
        for (int d = tid; d < HEAD_DIM; d += NWAVES * 32)
#pragma unroll
            for (int r = 0; r < GROUP; ++r)
                dst[2 * GROUP + r * HEAD_DIM + d] =
                    f0[r] * mO[0][r][d] + f1[r] * mO[1][r][d];
    } else {
        for (int d = tid; d < HEAD_DIM; d += NWAVES * 32)
#pragma unroll
            for (int r = 0; r < GROUP; ++r)
                dst[r * HEAD_DIM + d] =
                    (f0[r] * mO[0][r][d] + f1[r] * mO[1][r][d]) / Ls[r];
    }
}

#define DECLARE_LDS()                                  \
    __shared__ float sQ[16][132];                      \
    __shared__ float sK[NWAVES][2][16][132];           \
    __shared__ float sP[NWAVES][16 * 20];              \
    __shared__ unsigned long long sVp[NWAVES][2][16];  \
    __shared__ float mM[NWAVES][GROUP];                \
    __shared__ float mL[NWAVES][GROUP];                \
    __shared__ float mO[NWAVES][GROUP][HEAD_DIM];

// Fallback (no workspace): one block per (b, kvh), whole KV range.
__global__ __launch_bounds__(NWAVES * 32) void paged_attn_direct_kernel(
    const float* __restrict__ q, const float* __restrict__ kin,
    const float* __restrict__ vin, const float* __restrict__ kc,
    const float* __restrict__ vc, const int* __restrict__ slot_mapping,
    const int* __restrict__ page_indices, float* __restrict__ out)
{
    DECLARE_LDS();
    const int b   = blockIdx.x / NUM_KV_HEADS;
    const int kvh = blockIdx.x % NUM_KV_HEADS;
    float* dst = out + (size_t)b * NUM_HEADS * HEAD_DIM + kvh * GROUP * HEAD_DIM;
    attn_core(b, kvh, 0, NTILES, q, kin, vin, kc, vc, slot_mapping, page_indices,
              sQ, sK, sP, sVp, mM, mL, mO, dst, false);
}

// Pass 1: flash-decode partials. grid = (BS*NUM_KV_HEADS, NCHUNK).
__global__ __launch_bounds__(NWAVES * 32) void paged_attn_partial_kernel(
    const float* __restrict__ q, const float* __restrict__ kin,
    const float* __restrict__ vin, const float* __restrict__ kc,
    const float* __restrict__ vc, const int* __restrict__ slot_mapping,
    const int* __restrict__ page_indices, float* __restrict__ part)
{
    DECLARE_LDS();
    const int b     = blockIdx.x / NUM_KV_HEADS;
    const int kvh   = blockIdx.x % NUM_KV_HEADS;
    const int chunk = blockIdx.y;
    const int tpc   = NTILES / NCHUNK;
    float* dst = part + ((size_t)blockIdx.x * NCHUNK + chunk) * PART_STRIDE;
    attn_core(b, kvh, chunk * tpc, (chunk + 1) * tpc, q, kin, vin, kc, vc,
              slot_mapping, page_indices, sQ, sK, sP, sVp, mM, mL, mO, dst, true);
}

// Pass 2: combine NCHUNK partials per (b, kvh), normalize, store.
__global__ __launch_bounds__(128) void paged_attn_combine_kernel(
    const float* __restrict__ part, float* __restrict__ out)
{
    const int b   = blockIdx.x / NUM_KV_HEADS;
    const int kvh = blockIdx.x % NUM_KV_HEADS;
    const int d   = threadIdx.x;
    const float* p0 = part + (size_t)blockIdx.x * NCHUNK * PART_STRIDE;
#pragma unroll
    for (int r = 0; r < GROUP; ++r) {
        float gm = -INFINITY;
#pragma unroll
        for (int c = 0; c < NCHUNK; ++c) gm = fmaxf(gm, p0[c * PART_STRIDE + r]);
        float Ls = 0.f, val = 0.f;
#pragma unroll
        for (int c = 0; c < NCHUNK; ++c) {
            const float* p = p0 + c * PART_STRIDE;
            float f = __expf(p[r] - gm);
            Ls  += f * p[GROUP + r];
            val += f * p[2 * GROUP + r * HEAD_DIM + d];
        }
        out[(size_t)b * NUM_HEADS * HEAD_DIM + (kvh * GROUP + r) * HEAD_DIM + d] = val / Ls;
    }
}

extern "C" void kernel_launch(void* const* d_in, const int* in_sizes, int n_in,
                              void* d_out, int out_size, void* d_ws, size_t ws_size,
                              hipStream_t stream) {
    const float* q  = (const float*)d_in[0];
    const float* k  = (const float*)d_in[1];
    const float* v  = (const float*)d_in[2];
    const float* kc = (const float*)d_in[3];
    const float* vc = (const float*)d_in[4];
    const int* slot_mapping = (const int*)d_in[5];
    const int* page_indices = (const int*)d_in[6];
    float* out = (float*)d_out;

    const size_t part_bytes =
        (size_t)BS * NUM_KV_HEADS * NCHUNK * PART_STRIDE * sizeof(float);

    if (ws_size >= part_bytes && d_ws != nullptr) {
        // Flash-decode path: 1024 blocks (2048 waves) to saturate HBM.
        float* part = (float*)d_ws;
        dim3 grid1(BS * NUM_KV_HEADS, NCHUNK);
        paged_attn_partial_kernel<<<grid1, NWAVES * 32, 0, stream>>>(
            q, k, v, kc, vc, slot_mapping, page_indices, part);
        paged_attn_combine_kernel<<<dim3(BS * NUM_KV_HEADS), 128, 0, stream>>>(part, out);
    } else {
        paged_attn_direct_kernel<<<dim3(BS * NUM_KV_HEADS), NWAVES * 32, 0, stream>>>(
            q, k, v, kc, vc, slot_mapping, page_indices, out);
    }
}